// one_hot_CrossEntropy_2010044695292
// MI455X (gfx1250) — compile-verified
//
#include <hip/hip_runtime.h>
#include <hip/hip_bf16.h>
#include <math.h>

#define CE_EPS 1e-8f

__device__ __forceinline__ unsigned to_lds_u32(const void* p) {
    // low 32 bits of a generic pointer into LDS == wave-relative LDS byte address
    return (unsigned)(unsigned long long)(uintptr_t)p;
}

// ---------------------------------------------------------------------------
// Specialized kernel: C == 32000.  One block (320 threads = 10 wave32) per row.
// Row = 8000 float4 = 5 chunks x 1600 float4; each thread owns 5 float4/chunk.
// Async global->LDS double-buffered pipeline (ASYNCcnt), online softmax.
// ---------------------------------------------------------------------------
__launch_bounds__(320)
__global__ void ce_row_kernel_32000(const float* __restrict__ x,
                                    const int*   __restrict__ y,
                                    float*       __restrict__ row_loss) {
    constexpr int THREADS   = 320;
    constexpr int CHUNK_F4  = 1600;   // float4 per chunk
    constexpr int PER_TH    = 5;      // float4 per thread per chunk

    __shared__ float4 buf[2 * CHUNK_F4];   // 51200 B double buffer
    __shared__ float  wm[10], wsum[10];

    const int tid = threadIdx.x;
    const int row = blockIdx.x;
    const float* rowbase = x + (size_t)row * 32000;
    const unsigned lds0 = to_lds_u32(&buf[0]);

    // Issue one chunk's async copies: 5 x b128 per lane (5 instr per wave).
    auto issue = [&](int chunk, int bufsel) {
#pragma unroll
        for (int j = 0; j < PER_TH; ++j) {
            const int slot = j * THREADS + tid;
            const unsigned laddr = lds0 + (unsigned)(bufsel * CHUNK_F4 + slot) * 16u;
            const unsigned goff  = (unsigned)(chunk * CHUNK_F4 + slot) * 16u;
            asm volatile("global_load_async_to_lds_b128 %0, %1, %2 offset:0"
                         :: "v"(laddr), "v"(goff), "s"(rowbase)
                         : "memory");
        }
    };

    float m = -INFINITY, s = 0.f;

    // Consume one chunk from LDS: online max / sum-of-exp update.
    auto consume = [&](int bufsel) {
#pragma unroll
        for (int j = 0; j < PER_TH; ++j) {
            const float4 v = buf[bufsel * CHUNK_F4 + j * THREADS + tid];
            const float cm = fmaxf(fmaxf(v.x, v.y), fmaxf(v.z, v.w));
            if (cm > m) { s *= __expf(m - cm); m = cm; }
            s += __expf(v.x - m) + __expf(v.y - m) +
                 __expf(v.z - m) + __expf(v.w - m);
        }
    };

    // Double-buffered pipeline over 5 chunks. Per wave: 5 asyncs per chunk,
    // asyncs complete in order, so "asynccnt <= 5" == "oldest chunk landed".
    issue(0, 0);
    issue(1, 1);

    asm volatile("s_wait_asynccnt 5" ::: "memory");
    consume(0);
    asm volatile("s_wait_dscnt 0" ::: "memory");   // WAR: buf0 reads done
    issue(2, 0);

    asm volatile("s_wait_asynccnt 5" ::: "memory");
    consume(1);
    asm volatile("s_wait_dscnt 0" ::: "memory");
    issue(3, 1);

    asm volatile("s_wait_asynccnt 5" ::: "memory");
    consume(0);
    asm volatile("s_wait_dscnt 0" ::: "memory");
    issue(4, 0);

    asm volatile("s_wait_asynccnt 5" ::: "memory");
    consume(1);

    asm volatile("s_wait_asynccnt 0" ::: "memory");
    consume(0);

    // wave32 (m, s) merge
#pragma unroll
    for (int off = 16; off > 0; off >>= 1) {
        const float om = __shfl_xor(m, off, 32);
        const float os = __shfl_xor(s, off, 32);
        const float nm = fmaxf(m, om);
        s = s * __expf(m - nm) + os * __expf(om - nm);
        m = nm;
    }
    const int wave = tid >> 5, lane = tid & 31;
    if (lane == 0) { wm[wave] = m; wsum[wave] = s; }
    __syncthreads();

    if (tid == 0) {
        float M = wm[0], S = wsum[0];
#pragma unroll
        for (int w = 1; w < 10; ++w) {
            const float om = wm[w], os = wsum[w];
            const float nm = fmaxf(M, om);
            S = S * __expf(M - nm) + os * __expf(om - nm);
            M = nm;
        }
        const float logZ = M + __logf(S);
        const int   label = y[row];
        const float xy = rowbase[label];
        const float p  = __expf(xy - logZ);
        row_loss[row] = -__logf(p + CE_EPS);
    }
}

// ---------------------------------------------------------------------------
// Generic fallback for arbitrary C (plain streaming loads).
// ---------------------------------------------------------------------------
__launch_bounds__(256)
__global__ void ce_row_generic(const float* __restrict__ x,
                               const int*   __restrict__ y,
                               float*       __restrict__ row_loss, int C) {
    __shared__ float wm[8], wsum[8];
    const int tid = threadIdx.x;
    const int row = blockIdx.x;
    const float* rowbase = x + (size_t)row * C;

    float m = -INFINITY, s = 0.f;
    for (int i = tid; i < C; i += 256) {
        const float v = rowbase[i];
        if (v > m) { s *= __expf(m - v); m = v; }
        s += __expf(v - m);
    }
#pragma unroll
    for (int off = 16; off > 0; off >>= 1) {
        const float om = __shfl_xor(m, off, 32);
        const float os = __shfl_xor(s, off, 32);
        const float nm = fmaxf(m, om);
        s = s * __expf(m - nm) + os * __expf(om - nm);
        m = nm;
    }
    const int wave = tid >> 5, lane = tid & 31;
    if (lane == 0) { wm[wave] = m; wsum[wave] = s; }
    __syncthreads();
    if (tid == 0) {
        float M = wm[0], S = wsum[0];
#pragma unroll
        for (int w = 1; w < 8; ++w) {
            const float om = wm[w], os = wsum[w];
            const float nm = fmaxf(M, om);
            S = S * __expf(M - nm) + os * __expf(om - nm);
            M = nm;
        }
        const float logZ = M + __logf(S);
        const float xy = rowbase[y[row]];
        row_loss[row] = -__logf(__expf(xy - logZ) + CE_EPS);
    }
}

// ---------------------------------------------------------------------------
// Deterministic mean of N row losses -> scalar.
// ---------------------------------------------------------------------------
__launch_bounds__(256)
__global__ void ce_mean_kernel(const float* __restrict__ v,
                               float* __restrict__ out, int n) {
    __shared__ float acc[8];
    float s = 0.f;
    for (int i = threadIdx.x; i < n; i += 256) s += v[i];
#pragma unroll
    for (int off = 16; off > 0; off >>= 1) s += __shfl_xor(s, off, 32);
    const int wave = threadIdx.x >> 5, lane = threadIdx.x & 31;
    if (lane == 0) acc[wave] = s;
    __syncthreads();
    if (threadIdx.x == 0) {
        float t = 0.f;
#pragma unroll
        for (int w = 0; w < 8; ++w) t += acc[w];
        out[0] = t / (float)n;
    }
}

extern "C" void kernel_launch(void* const* d_in, const int* in_sizes, int n_in,
                              void* d_out, int out_size, void* d_ws, size_t ws_size,
                              hipStream_t stream) {
    const float* x = (const float*)d_in[0];
    const int*   y = (const int*)d_in[1];
    float* out = (float*)d_out;
    float* row_loss = (float*)d_ws;

    const int N = in_sizes[1];
    const int C = in_sizes[0] / N;

    if (C == 32000) {
        ce_row_kernel_32000<<<N, 320, 0, stream>>>(x, y, row_loss);
    } else {
        ce_row_generic<<<N, 256, 0, stream>>>(x, y, row_loss, C);
    }
    ce_mean_kernel<<<1, 256, 0, stream>>>(row_loss, out, N);
}